// SupernovaFIXScores_26998164422906
// MI455X (gfx1250) — compile-verified
//
#include <hip/hip_runtime.h>
#include <stdint.h>

// ---------------------------------------------------------------------------
// SupernovaFIXScores on MI455X (gfx1250, wave32)
//
// B=32 batches -> one workgroup each, 384 threads = 12 waves.
// The five masked moments per (g,w) row are a GEMM
//   mask(192 x 4096) * features(4096 x 16-padded)  via V_WMMA_F32_16X16X4_F32,
// wave i owns M-tile rows [16i,16i+16), K stepped 4/WMMA, acc chained in C.
// Masks are 1-bit-per-sample words (group AND wavelength), features are
// synthesized branchlessly from an interleaved (t,f) LDS record (1x b128/step).
// ---------------------------------------------------------------------------

#define B_      32
#define S_      4096
#define G_      32
#define W_      6
#define NCHUNK_ 7
#define SIGMA_  1.0f
#define EPS_    1e-5f
#define NROWS_  (G_ * W_)        // 192
#define NWAVES_ 12
#define BLOCK_  (NWAVES_ * 32)   // 384
#define NW_     (S_ / 32)        // 128 mask words per row

typedef __attribute__((ext_vector_type(2))) float v2f;
typedef __attribute__((ext_vector_type(8))) float v8f;

__device__ __forceinline__ float wave_red_sum(float v) {
#pragma unroll
  for (int o = 16; o; o >>= 1) v += __shfl_xor(v, o, 32);
  return v;
}
__device__ __forceinline__ float wave_red_min(float v) {
#pragma unroll
  for (int o = 16; o; o >>= 1) v = fminf(v, __shfl_xor(v, o, 32));
  return v;
}
__device__ __forceinline__ float wave_red_max(float v) {
#pragma unroll
  for (int o = 16; o; o >>= 1) v = fmaxf(v, __shfl_xor(v, o, 32));
  return v;
}
__device__ __forceinline__ unsigned long long wave_red_or(unsigned long long v) {
#pragma unroll
  for (int o = 16; o; o >>= 1) v |= __shfl_xor(v, o, 32);
  return v;
}

__global__ __launch_bounds__(BLOCK_, 1)
void supernova_fix_scores_kernel(const float* __restrict__ past_values,       // (B,S,2) flux,err
                                 const float* __restrict__ past_time,         // (B,S,2) t,wl
                                 const unsigned char* __restrict__ groups,    // (B,G,S) bool
                                 float* __restrict__ out) {                   // (B,)
  __shared__ __align__(16) float sh_tf[2 * S_];    // interleaved (t,f): 32KB
  __shared__ float         sh_e[S_];               // err: 16KB
  __shared__ unsigned char sh_w[S_];               // wavelength index 0..5: 4KB
  __shared__ unsigned int  sh_gb[G_][NW_];         // group bitmasks: 16KB
  __shared__ unsigned int  sh_wb[W_][NW_];         // wavelength bitmasks: 3KB
  __shared__ float         sh_stats[NROWS_][6];    // cnt,sT,sF,sTT,sTF (+pad)
  __shared__ float         sh_pfp[NROWS_];         // percent_flux_pass (may be NaN)
  __shared__ float         sh_score[G_];
  __shared__ float         sh_red[NWAVES_];

  const int tid  = threadIdx.x;
  const int lane = tid & 31;
  const int wave = tid >> 5;
  const int b    = blockIdx.x;

  // ---------------- Stage 0a: load batch into LDS ----------------
  const float* pv  = past_values + (size_t)b * S_ * 2;
  const float* ptf = past_time   + (size_t)b * S_ * 2;
  for (int s = tid; s < S_; s += BLOCK_) {
    float t  = ptf[2 * s + 0];
    float wl = ptf[2 * s + 1];
    sh_tf[2 * s + 0] = t;
    sh_tf[2 * s + 1] = pv[2 * s + 0];
    sh_e[s]          = pv[2 * s + 1];
    int wi = 5;
    if      (wl == 3670.69f) wi = 0;
    else if (wl == 4826.85f) wi = 1;
    else if (wl == 6223.24f) wi = 2;
    else if (wl == 7545.98f) wi = 3;
    else if (wl == 8590.9f)  wi = 4;
    sh_w[s] = (unsigned char)wi;
  }
  const unsigned char* grp = groups + (size_t)b * G_ * S_;
  for (int widx = tid; widx < G_ * NW_; widx += BLOCK_) {
    int g = widx >> 7;                 // / NW_
    int j = widx & (NW_ - 1);
    const unsigned char* p = grp + (size_t)g * S_ + (size_t)j * 32;
    unsigned int bits = 0u;
#pragma unroll
    for (int k = 0; k < 32; ++k) bits |= (p[k] ? 1u : 0u) << k;
    sh_gb[g][j] = bits;
  }
  __syncthreads();

  // ---------------- Stage 0b: wavelength bitmasks from sh_w ----------------
  for (int widx = tid; widx < W_ * NW_; widx += BLOCK_) {
    int w = widx >> 7;
    int j = widx & (NW_ - 1);
    unsigned int bits = 0u;
#pragma unroll
    for (int k = 0; k < 32; ++k)
      bits |= (sh_w[j * 32 + k] == (unsigned char)w ? 1u : 0u) << k;
    sh_wb[w][j] = bits;
  }
  __syncthreads();

  // ---------------- Stage 1: WMMA moment GEMM (branchless) ----------------
  // A (16x4 f32): lanes 0-15 -> M=lane, VGPR0=K0 VGPR1=K1; lanes 16-31 -> K2,K3.
  // B (4x16 f32): lanes 0-15 -> N=lane, VGPR0=K0 VGPR1=K1; lanes 16-31 -> K2,K3.
  // C/D (16x16 f32): lane -> N=lane&15; VGPR r -> M = r + 8*(lane>>4).
  {
    const int m    = lane & 15;
    const int half = lane >> 4;
    const int row  = wave * 16 + m;      // (g,w) row of A for this lane
    const int gA   = row / W_;
    const int wvA  = row % W_;
    const int nB   = m;                  // feature column of B for this lane
    // 0/1 feature-selection coefficients (lane constants)
    const float c1  = (nB == 0) ? 1.0f : 0.0f;
    const float ct  = (nB == 1) ? 1.0f : 0.0f;
    const float cf  = (nB == 2) ? 1.0f : 0.0f;
    const float ctt = (nB == 3) ? 1.0f : 0.0f;
    const float ctf = (nB == 4) ? 1.0f : 0.0f;
    v8f acc = {};
    for (int blk = 0; blk < NW_; ++blk) {
      // one mask word covers 32 K-samples = 8 WMMAs
      const unsigned int mw = sh_gb[gA][blk] & sh_wb[wvA][blk];
#pragma unroll
      for (int it = 0; it < 8; ++it) {
        const int sb    = blk * 32 + it * 4 + 2 * half;   // even
        const int shift = it * 4 + 2 * half;
        const float4 rec = *((const float4*)&sh_tf[2 * sb]);  // t0,f0,t1,f1 (b128)
        v2f a, bm;
        a.x = (float)((mw >> shift) & 1u);
        a.y = (float)((mw >> (shift + 1)) & 1u);
        {
          const float tt = rec.x, ff = rec.y;
          float v = fmaf(tt, ct, c1);
          v = fmaf(ff, cf, v);
          v = fmaf(tt * tt, ctt, v);
          bm.x = fmaf(tt * ff, ctf, v);
        }
        {
          const float tt = rec.z, ff = rec.w;
          float v = fmaf(tt, ct, c1);
          v = fmaf(ff, cf, v);
          v = fmaf(tt * tt, ctt, v);
          bm.y = fmaf(tt * ff, ctf, v);
        }
        acc = __builtin_amdgcn_wmma_f32_16x16x4_f32(
            /*neg_a=*/false, a, /*neg_b=*/false, bm,
            /*c_mod=*/(short)0, acc, /*reuse_a=*/false, /*reuse_b=*/false);
      }
    }
#pragma unroll
    for (int r = 0; r < 8; ++r) {
      const int rowD = wave * 16 + r + 8 * half;
      if (nB < 5) sh_stats[rowD][nB] = acc[r];
    }
  }
  __syncthreads();

  // ---------------- Stage 2: regression + percent_flux_pass ----------------
  for (int rr = 0; rr < 16; ++rr) {
    const int row = wave * 16 + rr;
    const int g   = row / W_;
    const int wv  = row % W_;
    const float c   = sh_stats[row][0];
    const float sT  = sh_stats[row][1];
    const float sF  = sh_stats[row][2];
    const float sTT = sh_stats[row][3];
    const float sTF = sh_stats[row][4];
    const float cc  = fmaxf(c, EPS_);
    const float tm  = sT / cc;
    const float fm  = sF / cc;
    const float num = sTF - fm * sT - tm * sF + c * tm * fm;
    const float den = sTT - 2.0f * tm * sT + c * tm * tm;
    const float beta  = num / den;                 // 0/0 -> NaN, as in reference
    const float alpha = fm - tm * beta;
    float cnt = 0.0f;
    for (int blk = 0; blk < NW_; ++blk) {
      const int s = blk * 32 + lane;
      const unsigned int mw = sh_gb[g][blk] & sh_wb[wv][blk];   // broadcast word
      const float mk = (float)((mw >> lane) & 1u);
      const float t  = sh_tf[2 * s + 0];
      const float f  = sh_tf[2 * s + 1];
      const float pf = (fabsf(fmaf(t, beta, alpha) - f) <= SIGMA_ * sh_e[s]) ? 1.0f : 0.0f;
      cnt += mk * pf;
    }
    cnt = wave_red_sum(cnt);
    if (lane == 0) sh_pfp[row] = cnt / c;          // 0/0 -> NaN (matches jnp)
  }
  __syncthreads();

  // ---------------- Stage 3: chunk coverage + NaN-aware max over w ----------
  for (int g = wave; g < G_; g += NWAVES_) {
    // groups_t = t * groups: masked-out entries contribute 0 to min/max
    float lo = 3.4e38f, hi = -3.4e38f;
    for (int blk = 0; blk < NW_; ++blk) {
      const int s = blk * 32 + lane;
      const unsigned int bit = (sh_gb[g][blk] >> lane) & 1u;
      const float v = bit ? sh_tf[2 * s] : 0.0f;
      lo = fminf(lo, v);
      hi = fmaxf(hi, v);
    }
    lo = wave_red_min(lo);
    hi = wave_red_max(hi);
    const float step   = (hi - lo) * (1.0f / (float)NCHUNK_);
    const float window = 2.0f * step;
    unsigned long long bits = 0ull;   // 6 wavelengths x 7 chunks = 42 "any" flags
    for (int blk = 0; blk < NW_; ++blk) {
      const int s = blk * 32 + lane;
      const float tt = sh_tf[2 * s];
      const int   wv = sh_w[s];
#pragma unroll
      for (int cch = 0; cch < NCHUNK_; ++cch) {
        const float cs = lo + (float)cch * step;
        const float ce = fminf(cs + window, hi);
        if (cs <= tt && tt <= ce) bits |= 1ull << (wv * NCHUNK_ + cch);
      }
    }
    bits = wave_red_or(bits);
    if (lane == 0) {
      float best = -3.4e38f;
      bool  has_nan = false;
      for (int wv = 0; wv < W_; ++wv) {
        const float pcp = (float)__popcll((bits >> (wv * NCHUNK_)) & 0x7Full) *
                          (1.0f / (float)NCHUNK_);
        const float al = sh_pfp[g * W_ + wv] * pcp;
        if (al != al) has_nan = true;              // jnp.max propagates NaN
        best = fmaxf(best, al);
      }
      sh_score[g] = has_nan ? 0.0f : best;         // nan_to_num after max
    }
  }
  __syncthreads();

  // ---------------- Stage 4: group-weighted mean over samples ---------------
  float partial = 0.0f;
  for (int s = tid; s < S_; s += BLOCK_) {
    float numv = 0.0f, denv = 0.0f;
    const int ww = s >> 5, wb = s & 31;
#pragma unroll
    for (int g = 0; g < G_; ++g) {
      if ((sh_gb[g][ww] >> wb) & 1u) { numv += sh_score[g]; denv += 1.0f; }
    }
    partial += numv / fmaxf(denv, EPS_);
  }
  partial = wave_red_sum(partial);
  if (lane == 0) sh_red[wave] = partial;
  __syncthreads();
  if (tid == 0) {
    float tot = 0.0f;
#pragma unroll
    for (int i = 0; i < NWAVES_; ++i) tot += sh_red[i];
    out[b] = tot * (1.0f / (float)S_);
  }
}

extern "C" void kernel_launch(void* const* d_in, const int* in_sizes, int n_in,
                              void* d_out, int out_size, void* d_ws, size_t ws_size,
                              hipStream_t stream) {
  (void)in_sizes; (void)n_in; (void)out_size; (void)d_ws; (void)ws_size;
  const float*         past_values = (const float*)d_in[0];         // (B,S,2) f32
  const float*         past_time   = (const float*)d_in[1];         // (B,S,2) f32
  const unsigned char* groups      = (const unsigned char*)d_in[2]; // (B,G,S) bool
  float* out = (float*)d_out;                                       // (B,) f32
  supernova_fix_scores_kernel<<<dim3(B_), dim3(BLOCK_), 0, stream>>>(
      past_values, past_time, groups, out);
}